// Conv4D_59811714564736
// MI455X (gfx1250) — compile-verified
//
#include <hip/hip_runtime.h>

// CDNA5 (gfx1250) wave32 WMMA kernel for separable 4D conv:
//   out = conv2d_ab(K) ( conv2d_cd(K) ( x ) ), x: (8192, 8,8,8,8) f32, K: 3x3
// Per batch item: X (64x64) -> TMP = X*Wt (64x48) -> OUT = Wt^T*TMP (48x48),
// store OUT[0:36, 0:36]. GEMMs via v_wmma_f32_16x16x4_f32 (full f32 precision).
// Bandwidth-bound: ~176MB traffic @ 23.3 TB/s => ~7.6us floor. Global->LDS
// staging uses the CDNA5 async pipe (global_load_async_to_lds_b128, ASYNCcnt)
// so the weight-matrix build overlaps the DMA and no VGPR round-trip is paid.

typedef __attribute__((ext_vector_type(2))) float v2f;
typedef __attribute__((ext_vector_type(8))) float v8f;

#define XS 68   // LDS row stride for X (pad 64 -> 68: conflict-free b64 A-fetch)
#define WS 50   // LDS row stride for Wt / TMP (48 cols used, pad to 50)

__global__ __launch_bounds__(256) void conv4d_wmma_kernel(
    const float* __restrict__ x,
    const float* __restrict__ kern,
    float* __restrict__ out)
{
    __shared__ float Xs[64 * XS];   // 17408 B
    __shared__ float Wt[64 * WS];   // 12800 B  (Wt[(c,d)][(c',d')])
    __shared__ float Tp[64 * WS];   // 12800 B  (TMP[(a,b)][(c',d')])

    const int tid  = threadIdx.x;
    const int lane = tid & 31;
    const int wv   = tid >> 5;          // wave id, 0..7
    const int mrow = lane & 15;         // row-within-tile for A, col for B/C/D
    const int kb   = (lane >> 4) << 1;  // K-base per lane half: 0 or 2
    const size_t n = blockIdx.x;

    // ---- Stage 0a: async-copy X tile (64x64 f32, 16KB) HBM -> LDS ----
    // GVS mode: 64-bit SGPR base + 32-bit per-lane byte offset. LDS dest is
    // the low 32 bits of the flat shared pointer (= LDS offset per aperture).
    {
        const float* gbase = x + n * 4096;
        #pragma unroll
        for (int it = 0; it < 4; ++it) {
            int i4  = tid + it * 256;          // 0..1023 (16B chunks)
            int row = i4 >> 4;                 // 16 chunks per 64-col row
            int c4  = (i4 & 15) << 2;
            unsigned ldsoff = (unsigned)(size_t)(void*)&Xs[row * XS + c4];
            unsigned goff   = (unsigned)i4 << 4;   // byte offset
            asm volatile(
                "global_load_async_to_lds_b128 %0, %1, %2"
                :
                : "v"(ldsoff), "v"(goff), "s"(gbase)
                : "memory");
        }
    }

    // ---- Stage 0b: build banded weight matrix Wt (overlaps async DMA) ----
    // Wt[r=(c*8+d)][col=(c'*6+d')] = K[c-c', d-d'] if both offsets in [0,2]
    for (int i = tid; i < 64 * WS; i += 256) {
        int r = i / WS, c = i % WS;
        float v = 0.0f;
        if (c < 36) {
            int cs = r >> 3, d = r & 7;
            int cp = c / 6,  dp = c % 6;
            int kr = cs - cp, kc = d - dp;
            if (kr >= 0 && kr < 3 && kc >= 0 && kc < 3)
                v = kern[kr * 3 + kc];
        }
        Wt[i] = v;
    }

    // All of this wave's async LDS writes complete, then workgroup barrier.
    asm volatile("s_wait_asynccnt 0x0" ::: "memory");
    __syncthreads();

    // ---- Stage 1: TMP(64x48) = X(64x64) * Wt(64x48); 4x3 tiles over 8 waves ----
    for (int t = wv; t < 12; t += 8) {
        const int tm = t / 3, tn = t % 3;
        v8f acc = {};
        #pragma unroll
        for (int kc4 = 0; kc4 < 16; ++kc4) {
            const int k0 = (kc4 << 2) + kb;
            v2f a, b;
            const float* ap = &Xs[(tm * 16 + mrow) * XS + k0];
            a.x = ap[0];                       // A[m][k0]   (b64, conflict-free)
            a.y = ap[1];                       // A[m][k0+1]
            b.x = Wt[k0 * WS + tn * 16 + mrow];        // B[k0][col]
            b.y = Wt[(k0 + 1) * WS + tn * 16 + mrow];  // B[k0+1][col]
            acc = __builtin_amdgcn_wmma_f32_16x16x4_f32(
                false, a, false, b, (short)0, acc, false, false);
        }
        const int rbase = tm * 16 + ((lane >> 4) << 3);
        const int col   = tn * 16 + mrow;
        #pragma unroll
        for (int v = 0; v < 8; ++v)
            Tp[(rbase + v) * WS + col] = acc[v];
    }
    __syncthreads();

    // ---- Stage 2: OUT(48x48) = Wt^T(48x64) * TMP(64x48); 3x3 tiles ----
    float* const op = out + n * 1296;
    for (int t = wv; t < 9; t += 8) {
        const int tm = t / 3, tn = t % 3;
        v8f acc = {};
        #pragma unroll
        for (int kc4 = 0; kc4 < 16; ++kc4) {
            const int k0 = (kc4 << 2) + kb;
            v2f a, b;
            a.x = Wt[k0 * WS + tm * 16 + mrow];        // A[m][k] = Wt[k][m]
            a.y = Wt[(k0 + 1) * WS + tm * 16 + mrow];
            b.x = Tp[k0 * WS + tn * 16 + mrow];        // B[k][col]
            b.y = Tp[(k0 + 1) * WS + tn * 16 + mrow];
            acc = __builtin_amdgcn_wmma_f32_16x16x4_f32(
                false, a, false, b, (short)0, acc, false, false);
        }
        const int rbase = tm * 16 + ((lane >> 4) << 3);
        const int col   = tn * 16 + mrow;
        if (col < 36) {
            #pragma unroll
            for (int v = 0; v < 8; ++v) {
                const int R = rbase + v;
                if (R < 36) op[R * 36 + col] = acc[v];
            }
        }
    }
}

extern "C" void kernel_launch(void* const* d_in, const int* in_sizes, int n_in,
                              void* d_out, int out_size, void* d_ws, size_t ws_size,
                              hipStream_t stream) {
    const float* x    = (const float*)d_in[0];
    const float* kern = (const float*)d_in[1];
    float* out        = (float*)d_out;
    const int nbatch  = in_sizes[0] / 4096;   // 8192
    conv4d_wmma_kernel<<<nbatch, 256, 0, stream>>>(x, kern, out);
}